// transfHead_83623013253616
// MI455X (gfx1250) — compile-verified
//
#include <hip/hip_runtime.h>
#include <math.h>
#include <stdint.h>

// Problem constants (match reference)
#define IN_DIM 128
#define NH 8
#define OD 16
#define HD (NH * OD)   // 128 = per-matrix output width

typedef float v2f __attribute__((ext_vector_type(2)));
typedef float v8f __attribute__((ext_vector_type(8)));

// Only check for the device builtin in the DEVICE compile pass; amdgcn
// builtins are not declared in the host (x86) pass.
#if defined(__HIP_DEVICE_COMPILE__)
#if !__has_builtin(__builtin_amdgcn_wmma_f32_16x16x4_f32)
#error "device: __builtin_amdgcn_wmma_f32_16x16x4_f32 not available"
#endif
#endif

// ---------------------------------------------------------------------------
// Direct memory->LDS async copy (CDNA5 GLOBAL_LOAD_ASYNC_TO_LDS_B128,
// tracked by ASYNCcnt). Bypasses VGPRs; per-lane LDS byte offset comes from
// the low 32 bits of the generic pointer (generic LDS address = aperture
// high half + LDS offset per ISA 10.2). Inline asm keeps this portable
// across ROCm 7.2 / amdgpu-toolchain builtin-arity differences.
// ---------------------------------------------------------------------------
__device__ __forceinline__ void async_b128_to_lds(const float* gsrc,
                                                  const float* lds_dst)
{
    const unsigned lds_off = (unsigned)(uintptr_t)lds_dst;
    asm volatile("global_load_async_to_lds_b128 %0, %1, off"
                 :
                 : "v"(lds_off), "v"((const void*)gsrc)
                 : "memory");
}

__device__ __forceinline__ void wait_asynccnt0()
{
    asm volatile("s_wait_asynccnt 0x0" ::: "memory");
}

// ---------------------------------------------------------------------------
// Fused QKV projection: [N,128] x [128,128]x3 (+bias) via V_WMMA_F32_16X16X4_F32
// Block = 256 threads (8 waves). Each block owns one 16-row M tile.
// A tile (16x128 f32) staged in LDS with row stride 132 floats (528 B, 16B
// aligned, bank-conflict free: 4*m mod 64 distinct for m=0..15) via async
// b128 memory->LDS copies. Each wave computes 3 of the 24 N-tiles
// (8 Q + 8 K + 8 V), iterating K in steps of 4 (32 WMMAs per tile).
//
// WMMA f32 16x16x4 operand layout (ISA 7.12.2, wave32):
//   A (16x4):  lane l -> M = l%16 ; VGPR0 = A[M][kb + 2*(l/16)], VGPR1 = +1
//   B (4x16):  lane l -> N = l%16 ; VGPR0 = B[kb + 2*(l/16)][N], VGPR1 = +1
//   C/D (16x16): VGPR r, lanes 0-15 -> M=r, lanes 16-31 -> M=r+8 ; N = l%16
// ---------------------------------------------------------------------------
__global__ __launch_bounds__(256) void qkv_gemm_wmma(
    const float* __restrict__ state,
    const float* __restrict__ WQ, const float* __restrict__ bQ,
    const float* __restrict__ WK, const float* __restrict__ bK,
    const float* __restrict__ WV, const float* __restrict__ bV,
    float* __restrict__ Qb, float* __restrict__ Kb, float* __restrict__ Vb,
    int nNodes)
{
    __shared__ float As[16 * 132];

    const int m0  = blockIdx.x << 4;
    const int tid = threadIdx.x;

    // Async-stage the 16x128 A tile into LDS: 512 b128 chunks, 2 per thread.
    // Rows are clamped (never triggers for N = multiple of 16); clamped rows
    // only feed output rows that are never stored.
    for (int chunk = tid; chunk < 512; chunk += 256) {
        const int r  = chunk >> 5;          // tile row 0..15
        const int c4 = (chunk & 31) << 2;   // starting float col, step 4
        int m = m0 + r;
        if (m >= nNodes) m = nNodes - 1;
        async_b128_to_lds(state + (size_t)m * IN_DIM + c4,
                          As + r * 132 + c4);
    }
    wait_asynccnt0();     // each wave drains its own ASYNCcnt
    __syncthreads();      // then cross-wave visibility

    const int wave = tid >> 5;
    const int lane = tid & 31;
    const int mrow = lane & 15;          // A-fragment row
    const int koff = (lane >> 4) << 1;   // 0 for lanes 0-15, 2 for lanes 16-31

    for (int i = 0; i < 3; ++i) {
        const int nt  = wave * 3 + i;          // 0..23
        const int mat = nt >> 3;               // 0=Q, 1=K, 2=V
        const int ncol = ((nt & 7) << 4) + (lane & 15);  // column in [0,128)

        const float* __restrict__ W    = (mat == 0) ? WQ : (mat == 1) ? WK : WV;
        const float* __restrict__ bias = (mat == 0) ? bQ : (mat == 1) ? bK : bV;
        float* __restrict__ outb       = (mat == 0) ? Qb : (mat == 1) ? Kb : Vb;

        v8f c = {0.f, 0.f, 0.f, 0.f, 0.f, 0.f, 0.f, 0.f};

#pragma unroll 8
        for (int kb = 0; kb < IN_DIM; kb += 4) {
            v2f a, b;
            a.x = As[mrow * 132 + kb + koff];
            a.y = As[mrow * 132 + kb + koff + 1];
            b.x = W[(size_t)(kb + koff) * HD + ncol];
            b.y = W[(size_t)(kb + koff + 1) * HD + ncol];
            // (neg_a, A, neg_b, B, c_mod, C, reuse_a, reuse_b)
            c = __builtin_amdgcn_wmma_f32_16x16x4_f32(
                    false, a, false, b, (short)0, c, false, false);
        }

        const float bv    = bias[ncol];
        const int   mbase = m0 + ((lane >> 4) << 3);  // lanes 16-31 hold M = r+8
#pragma unroll
        for (int r = 0; r < 8; ++r) {
            const int m = mbase + r;
            if (m < nNodes) outb[(size_t)m * HD + ncol] = c[r] + bv;
        }
    }
}

// ---------------------------------------------------------------------------
// Edge kernel: one wave32 per edge. Lane l owns features 4l..4l+3 of the
// 128-wide row (head h = l>>2). float4 (b128) gathers of K[src], Q[dst],
// V[src] (all L2-resident: QKV working set = 78 MB < 192 MB L2); head-wise
// dot reduced across the 4 lanes of each head with two shfl_xor
// (ds-permute); clamped exp via v_exp_f32; scatter via
// global_atomic_add_f32 into out[dst] and z[dst].
// ---------------------------------------------------------------------------
__global__ __launch_bounds__(256) void edge_attn(
    const float* __restrict__ Qb, const float* __restrict__ Kb,
    const float* __restrict__ Vb,
    const int* __restrict__ src, const int* __restrict__ dst,
    float* __restrict__ out, float* __restrict__ z, int nEdges)
{
    const int lane = threadIdx.x & 31;
    const int e    = (blockIdx.x << 3) + (threadIdx.x >> 5);
    if (e >= nEdges) return;   // uniform per wave

    const int s = src[e];
    const int d = dst[e];

    const float4 kv = *(const float4*)(Kb + (size_t)s * HD + lane * 4);
    const float4 qv = *(const float4*)(Qb + (size_t)d * HD + lane * 4);
    const float4 vv = *(const float4*)(Vb + (size_t)s * HD + lane * 4);

    float p = kv.x * qv.x + kv.y * qv.y + kv.z * qv.z + kv.w * qv.w;
    // reduce across the 4 lanes belonging to this head
    p += __shfl_xor(p, 1);
    p += __shfl_xor(p, 2);

    float sc = p * 0.25f;                        // 1/sqrt(OUT_DIM=16)
    sc = fminf(fmaxf(sc, -5.0f), 5.0f);
    const float w = __expf(sc);

    float* o = out + (size_t)d * HD + lane * 4;
    atomicAdd(o + 0, w * vv.x);
    atomicAdd(o + 1, w * vv.y);
    atomicAdd(o + 2, w * vv.z);
    atomicAdd(o + 3, w * vv.w);

    if ((lane & 3) == 0)
        atomicAdd(z + (size_t)d * NH + (lane >> 2), w);
}

// ---------------------------------------------------------------------------
__global__ void zero_two(float* __restrict__ a, int na,
                         float* __restrict__ b, int nb)
{
    const int i = blockIdx.x * blockDim.x + threadIdx.x;
    if (i < na)            a[i] = 0.0f;
    else if (i < na + nb)  b[i - na] = 0.0f;
}

__global__ void normalize_out(float* __restrict__ out,
                              const float* __restrict__ z, int total)
{
    const int i = blockIdx.x * blockDim.x + threadIdx.x;
    if (i < total) {
        const int node = i >> 7;          // /128
        const int h    = (i >> 4) & 7;    // head
        out[i] /= z[node * NH + h];
    }
}

// ---------------------------------------------------------------------------
extern "C" void kernel_launch(void* const* d_in, const int* in_sizes, int n_in,
                              void* d_out, int out_size, void* d_ws, size_t ws_size,
                              hipStream_t stream)
{
    const float* state = (const float*)d_in[0];
    const int*   src   = (const int*)d_in[1];
    const int*   dst   = (const int*)d_in[2];
    const float* WQ    = (const float*)d_in[3];
    const float* bQ    = (const float*)d_in[4];
    const float* WK    = (const float*)d_in[5];
    const float* bK    = (const float*)d_in[6];
    const float* WV    = (const float*)d_in[7];
    const float* bV    = (const float*)d_in[8];

    const int nNodes = in_sizes[0] / IN_DIM;
    const int nEdges = in_sizes[1];

    float* out = (float*)d_out;

    // Workspace layout: Q | K | V | z  (3*N*128 + N*8 floats ~ 78.4 MB)
    float* Qb = (float*)d_ws;
    float* Kb = Qb + (size_t)nNodes * HD;
    float* Vb = Kb + (size_t)nNodes * HD;
    float* zb = Vb + (size_t)nNodes * HD;

    const int totOut = nNodes * HD;
    const int totZ   = nNodes * NH;

    zero_two<<<(totOut + totZ + 255) / 256, 256, 0, stream>>>(out, totOut,
                                                              zb, totZ);

    qkv_gemm_wmma<<<(nNodes + 15) / 16, 256, 0, stream>>>(
        state, WQ, bQ, WK, bK, WV, bV, Qb, Kb, Vb, nNodes);

    edge_attn<<<(nEdges + 7) / 8, 256, 0, stream>>>(
        Qb, Kb, Vb, src, dst, out, zb, nEdges);

    normalize_out<<<(totOut + 255) / 256, 256, 0, stream>>>(out, zb, totOut);
}